// WaveletTransform_352187318317
// MI455X (gfx1250) — compile-verified
//
#include <hip/hip_runtime.h>

// Fused Haar + bilinear-2x-upsample + concat: x:(16,3,512,512) f32 -> out:(16,15,512,512) f32.
// Pure bandwidth problem: 50 MB read + 252 MB write ~= 302 MB -> ~13 us floor @ 23.3 TB/s.
// CDNA5 paths: async global->LDS b128 staging (ASYNCcnt) + nontemporal b128 stores.
//
// Block = 256 threads, covers a 16x32 tile of subband coords = 32x64 output pixels x 5 bands.
// Halo x-values outside the image are never consumed (subband indices are clamped at the
// read stage), so edge tiles only need in-bounds *addresses* -> clamp b128 address, keep align.

#define TILE_H 16
#define TILE_W 32
#define XROWS  36          // 2*TILE_H + 4
#define XCOLS  72          // 2*TILE_W + 8 (origin at 2*jw0-4 for 16B alignment)
#define XS     76          // padded stride, 76*4 = 304 B = 19*16 -> rows stay 16B aligned
#define SROWS  18          // TILE_H + 2
#define SCOLS  34          // TILE_W + 2
#define SS     35          // padded stride

#if defined(__has_builtin)
#if __has_builtin(__builtin_amdgcn_global_load_async_to_lds_b128)
#define HAVE_ASYNC_B128 1
#endif
#if __has_builtin(__builtin_amdgcn_global_load_async_to_lds_b32)
#define HAVE_ASYNC_B32 1
#endif
#if __has_builtin(__builtin_amdgcn_s_wait_asynccnt)
#define HAVE_WAIT_ASYNC 1
#endif
#endif

typedef float v4f __attribute__((ext_vector_type(4)));
typedef int   v4i __attribute__((vector_size(4 * sizeof(int))));
typedef __attribute__((address_space(1))) int g_int;   // global (AS1) scalar
typedef __attribute__((address_space(3))) int l_int;   // LDS (AS3) scalar
typedef __attribute__((address_space(1))) v4i g_v4i;   // global (AS1) 16B vector
typedef __attribute__((address_space(3))) v4i l_v4i;   // LDS (AS3) 16B vector

__global__ __launch_bounds__(256) void WaveletTransform_352187318317_kernel(
    const float* __restrict__ x, float* __restrict__ out) {
  __shared__ float Xs[XROWS * XS];
  __shared__ float Sb[4][SROWS * SS];

  const int tile_x = blockIdx.x;          // 0..7
  const int tile_y = blockIdx.y;          // 0..15
  const int bc     = blockIdx.z;          // 0..47  (b*3 + c)
  const int b = bc / 3;
  const int c = bc - 3 * b;

  const int jh0  = tile_y * TILE_H;       // subband row origin
  const int jw0  = tile_x * TILE_W;       // subband col origin
  const int row0 = 2 * jh0 - 2;           // x tile row origin (may be <0 at edges)
  const int col0 = 2 * jw0 - 4;           // x tile col origin, multiple of 4 (16B aligned)

  const int tid = threadIdx.x;
  const float* __restrict__ xin = x + (size_t)bc * (512 * 512);

  // ---- Stage 1: async-stage 36x72 halo'd x tile into LDS, 16B chunks ----
  // 36 rows x 18 b128 = 648 chunks; addresses clamped in-bounds (halo values unused).
  for (int e = tid; e < XROWS * (XCOLS / 4); e += 256) {
    const int r = e / (XCOLS / 4);
    const int i = e - r * (XCOLS / 4);
    int gr = row0 + r;      gr = gr < 0 ? 0 : (gr > 511 ? 511 : gr);
    int gc = col0 + 4 * i;  gc = gc < 0 ? 0 : (gc > 508 ? 508 : gc);
    const float* src = xin + gr * 512 + gc;
    float* dst = &Xs[r * XS + 4 * i];
#if defined(HAVE_ASYNC_B128)
    __builtin_amdgcn_global_load_async_to_lds_b128(
        (g_v4i*)(void*)src, (l_v4i*)dst, 0, 0);
#elif defined(HAVE_ASYNC_B32)
    #pragma unroll
    for (int w = 0; w < 4; ++w)
      __builtin_amdgcn_global_load_async_to_lds_b32(
          (g_int*)(void*)(src + w), (l_int*)(dst + w), 0, 0);
#else
    #pragma unroll
    for (int w = 0; w < 4; ++w) dst[w] = src[w];
#endif
  }
#if defined(HAVE_ASYNC_B128) || defined(HAVE_ASYNC_B32)
#if defined(HAVE_WAIT_ASYNC)
  __builtin_amdgcn_s_wait_asynccnt(0);
#else
  asm volatile("s_wait_asynccnt 0" ::: "memory");
#endif
#endif
  __syncthreads();

  // ---- Stage 2: 18x34 subband tiles (halo 1) from the x tile ----
  for (int e = tid; e < SROWS * SCOLS; e += 256) {
    const int sr = e / SCOLS;
    const int sc = e - sr * SCOLS;
    const int xr = 2 * sr, xc = 2 * sc + 2;     // local x coords
    const float a  = Xs[xr * XS + xc];
    const float bb = Xs[xr * XS + xc + 1];
    const float cc = Xs[(xr + 1) * XS + xc];
    const float dd = Xs[(xr + 1) * XS + xc + 1];
    Sb[0][sr * SS + sc] = (a + bb + cc + dd) * 0.5f;   // LL
    Sb[1][sr * SS + sc] = (a + bb - cc - dd) * 0.5f;   // LH
    Sb[2][sr * SS + sc] = (a - bb + cc - dd) * 0.5f;   // HL
    Sb[3][sr * SS + sc] = (a - bb - cc + dd) * 0.5f;   // HH
  }
  __syncthreads();

  // ---- Stage 3: each thread -> 2 subband cols = 4 output cols x 2 rows x 5 bands ----
  const int ty = tid >> 4, tx = tid & 15;     // 16 x 16 threads
  const int jh = jh0 + ty;                    // subband row
  const int q0 = jw0 + 2 * tx;                // first subband col (q1 = q0+1)
  const int lr  = ty + 1;                     // center row in subband tile
  const int lrm = (jh == 0)   ? lr : lr - 1;  // clamp == JAX edge renormalization
  const int lrp = (jh == 255) ? lr : lr + 1;
  const int c0 = 2 * tx + 1;                  // q0 local col
  const int c1 = c0 + 1;                      // q1 local col
  const int cm = (q0 == 0)       ? c0 : c0 - 1;
  const int cp = (q0 + 1 == 255) ? c1 : c1 + 1;

  const int h0 = 2 * jh;
  const int w0 = 2 * q0;                      // multiple of 4 -> 16B aligned stores
  float* __restrict__ obase = out + ((size_t)b * 15 + (size_t)c * 5) * (512 * 512);

  {  // band 0: passthrough of x straight from LDS (16B-aligned v4f reads)
    const int xr = 2 * ty + 2, xc = 4 * tx + 4;
    v4f t  = *(const v4f*)&Xs[xr * XS + xc];
    v4f bo = *(const v4f*)&Xs[(xr + 1) * XS + xc];
    __builtin_nontemporal_store(t,  (v4f*)(obase + (size_t)h0 * 512 + w0));
    __builtin_nontemporal_store(bo, (v4f*)(obase + (size_t)(h0 + 1) * 512 + w0));
  }

#pragma unroll
  for (int k = 0; k < 4; ++k) {
    const float* S = Sb[k];
    const float smm = S[lrm * SS + cm], sm0 = S[lrm * SS + c0], sm1 = S[lrm * SS + c1], smp = S[lrm * SS + cp];
    const float scm = S[lr  * SS + cm], sc0 = S[lr  * SS + c0], sc1 = S[lr  * SS + c1], scp = S[lr  * SS + cp];
    const float spm = S[lrp * SS + cm], sp0 = S[lrp * SS + c0], sp1 = S[lrp * SS + c1], spp = S[lrp * SS + cp];
    // vertical: out row 2jh = 0.25*up + 0.75*center ; 2jh+1 = 0.75*center + 0.25*down
    const float tm = 0.75f * scm + 0.25f * smm, t0 = 0.75f * sc0 + 0.25f * sm0;
    const float t1 = 0.75f * sc1 + 0.25f * sm1, tp = 0.75f * scp + 0.25f * smp;
    const float bm = 0.75f * scm + 0.25f * spm, b0 = 0.75f * sc0 + 0.25f * sp0;
    const float b1 = 0.75f * sc1 + 0.25f * sp1, bp = 0.75f * scp + 0.25f * spp;
    // horizontal: cols 2q0, 2q0+1, 2q1, 2q1+1
    v4f t  = { 0.75f * t0 + 0.25f * tm, 0.75f * t0 + 0.25f * t1,
               0.75f * t1 + 0.25f * t0, 0.75f * t1 + 0.25f * tp };
    v4f bo = { 0.75f * b0 + 0.25f * bm, 0.75f * b0 + 0.25f * b1,
               0.75f * b1 + 0.25f * b0, 0.75f * b1 + 0.25f * bp };
    float* __restrict__ ob = obase + (size_t)(k + 1) * (512 * 512);
    __builtin_nontemporal_store(t,  (v4f*)(ob + (size_t)h0 * 512 + w0));
    __builtin_nontemporal_store(bo, (v4f*)(ob + (size_t)(h0 + 1) * 512 + w0));
  }
}

extern "C" void kernel_launch(void* const* d_in, const int* in_sizes, int n_in,
                              void* d_out, int out_size, void* d_ws, size_t ws_size,
                              hipStream_t stream) {
  (void)in_sizes; (void)n_in; (void)out_size; (void)d_ws; (void)ws_size;
  const float* x = (const float*)d_in[0];
  float* out = (float*)d_out;
  dim3 grid(256 / TILE_W, 256 / TILE_H, 16 * 3);  // (8,16,48)
  dim3 block(256);
  WaveletTransform_352187318317_kernel<<<grid, block, 0, stream>>>(x, out);
}